// Set2SetPooling_47974784696373
// MI455X (gfx1250) — compile-verified
//
#include <hip/hip_runtime.h>

#define UNITS      256
#define STEPS      3
#define NUM_NODES  262144
#define NUM_GRAPHS 2048
#define GATES      (4 * UNITS)   // 1024
#define QSTAR      (2 * UNITS)   // 512

typedef float v2f __attribute__((ext_vector_type(2)));
typedef float v8f __attribute__((ext_vector_type(8)));

__device__ __forceinline__ float sigmoidf_(float x) {
    return 1.0f / (1.0f + __expf(-x));
}

// ------------------------------------------------------------------
// zero fill (grid-stride)
// ------------------------------------------------------------------
__global__ void zero_kernel(float* __restrict__ p, int n) {
    int i = blockIdx.x * blockDim.x + threadIdx.x;
    int stride = gridDim.x * blockDim.x;
    for (; i < n; i += stride) p[i] = 0.0f;
}

// ------------------------------------------------------------------
// segment starts: graph_indices is sorted; lower_bound per graph
// ------------------------------------------------------------------
__global__ void seg_starts_kernel(const int* __restrict__ seg,
                                  int* __restrict__ starts) {
    int g = blockIdx.x * blockDim.x + threadIdx.x;
    if (g > NUM_GRAPHS) return;
    if (g == NUM_GRAPHS) { starts[g] = NUM_NODES; return; }
    int lo = 0, hi = NUM_NODES;
    while (lo < hi) {
        int mid = (lo + hi) >> 1;
        if (seg[mid] < g) lo = mid + 1; else hi = mid;
    }
    starts[g] = lo;
}

// ------------------------------------------------------------------
// gates[G,1024] = [q_star | h] (G x 768) @ [W_i ; W_h] (768 x 1024)
// one wave per 16x16 output tile; V_WMMA_F32_16X16X4_F32, K-loop of 192
// A 16x4 f32 layout: lanes 0-15 -> M=lane, regs = K0,K1; lanes 16-31 -> K2,K3
// B 4x16 f32:        reg r -> K=r (lanes 0-15) / K=r+2 (lanes 16-31), N=lane&15
// C/D 16x16 f32:     reg r -> M = r + 8*(lane>>4), N = lane&15
// ------------------------------------------------------------------
__global__ void gates_gemm_kernel(const float* __restrict__ q_star,
                                  const float* __restrict__ h,
                                  const float* __restrict__ W_i,
                                  const float* __restrict__ W_h,
                                  float* __restrict__ gates) {
    const int nt   = blockIdx.x;        // 0..63  (N tiles of 1024)
    const int mt   = blockIdx.y;        // 0..127 (M tiles of 2048)
    const int lane = threadIdx.x;       // wave32, EXEC all-1s
    const int half = lane >> 4;         // 0 -> K{0,1}, 1 -> K{2,3}
    const int l15  = lane & 15;
    const int rowA = mt * 16 + l15;
    const int colB = nt * 16 + l15;

    v8f acc = {};
    const float* aq = q_star + (size_t)rowA * QSTAR;
    const float* ah = h      + (size_t)rowA * UNITS;

#pragma unroll 4
    for (int k = 0; k < QSTAR; k += 4) {        // 128 WMMAs over W_i
        const int ka = k + half * 2;
        v2f a; a.x = aq[ka];  a.y = aq[ka + 1];
        v2f b; b.x = W_i[(size_t)(ka    ) * GATES + colB];
               b.y = W_i[(size_t)(ka + 1) * GATES + colB];
        acc = __builtin_amdgcn_wmma_f32_16x16x4_f32(
                  false, a, false, b, (short)0, acc, false, false);
    }
#pragma unroll 4
    for (int k = 0; k < UNITS; k += 4) {        // 64 WMMAs over W_h
        const int ka = k + half * 2;
        v2f a; a.x = ah[ka];  a.y = ah[ka + 1];
        v2f b; b.x = W_h[(size_t)(ka    ) * GATES + colB];
               b.y = W_h[(size_t)(ka + 1) * GATES + colB];
        acc = __builtin_amdgcn_wmma_f32_16x16x4_f32(
                  false, a, false, b, (short)0, acc, false, false);
    }

    const int rowD = mt * 16 + half * 8;
#pragma unroll
    for (int r = 0; r < 8; ++r)
        gates[(size_t)(rowD + r) * GATES + nt * 16 + l15] = acc[r];
}

// ------------------------------------------------------------------
// LSTM elementwise: i,f,g,o split; writes c, h, and q_star[:, :256] = h
// ------------------------------------------------------------------
__global__ void lstm_cell_kernel(const float* __restrict__ gates,
                                 const float* __restrict__ bias,
                                 float* __restrict__ h,
                                 float* __restrict__ c,
                                 float* __restrict__ q_star) {
    const int idx = blockIdx.x * blockDim.x + threadIdx.x; // 0 .. G*U-1
    const int g   = idx >> 8;
    const int u   = idx & (UNITS - 1);
    const float* gr = gates + (size_t)g * GATES;

    float ig = sigmoidf_(gr[u]             + bias[u]);
    float fg = sigmoidf_(gr[UNITS + u]     + bias[UNITS + u]);
    float gg = tanhf    (gr[2 * UNITS + u] + bias[2 * UNITS + u]);
    float og = sigmoidf_(gr[3 * UNITS + u] + bias[3 * UNITS + u]);

    float cn = fg * c[idx] + ig * gg;
    float hn = og * tanhf(cn);
    c[idx] = cn;
    h[idx] = hn;
    q_star[(size_t)g * QSTAR + u] = hn;
}

// ------------------------------------------------------------------
// Attention per graph: one 256-thread block (8 wave32) per graph.
// SINGLE streaming pass over this graph's node rows (online softmax):
// per wave keep running (m, z, r[256]); on a new max (wave-uniform
// branch: the reduced logit d is identical in all lanes) rescale z and
// r by exp(m_old - m_new). Cross-wave merge is a fixed-order LDS tree
// with exp(m_w - m_block) correction -> deterministic.
// Halves inputs traffic vs. 2-pass softmax: 256 MB/step instead of 512.
// Writes q_star[:, 256:512] = r / z.
// ------------------------------------------------------------------
__global__ void attention_kernel(const float* __restrict__ inputs,
                                 const float* __restrict__ h,
                                 const int* __restrict__ starts,
                                 float* __restrict__ q_star) {
    __shared__ float q_sh[UNITS];
    __shared__ float rp[8 * UNITS];     // per-wave partial r
    __shared__ float m_sh[8];
    __shared__ float z_sh[8];

    const int g    = blockIdx.x;
    const int tid  = threadIdx.x;       // 256
    const int wave = tid >> 5;
    const int lane = tid & 31;
    const int base = lane * 8;          // 8 features per lane

    q_sh[tid] = h[(size_t)g * UNITS + tid];
    __syncthreads();

    const float q0 = q_sh[base + 0], q1 = q_sh[base + 1];
    const float q2 = q_sh[base + 2], q3 = q_sh[base + 3];
    const float q4 = q_sh[base + 4], q5 = q_sh[base + 5];
    const float q6 = q_sh[base + 6], q7 = q_sh[base + 7];

    const int s = starts[g];
    const int e = starts[g + 1];

    // ---- single pass: online softmax + weighted accumulation ------
    float m = -__builtin_inff();
    float z = 0.0f;
    float racc[8] = {0.f, 0.f, 0.f, 0.f, 0.f, 0.f, 0.f, 0.f};

    for (int n = s + wave; n < e; n += 8) {
        const float* row = inputs + (size_t)n * UNITS + base;
        float4 x0 = *(const float4*)(row);
        float4 x1 = *(const float4*)(row + 4);
        float d = x0.x * q0 + x0.y * q1 + x0.z * q2 + x0.w * q3
                + x1.x * q4 + x1.y * q5 + x1.z * q6 + x1.w * q7;
#pragma unroll
        for (int off = 16; off > 0; off >>= 1)
            d += __shfl_xor(d, off, 32);

        if (d > m) {                         // wave-uniform, rare
            const float corr = __expf(m - d);  // exp(-inf)=0 on first node
            z *= corr;
#pragma unroll
            for (int j = 0; j < 8; ++j) racc[j] *= corr;
            m = d;
        }
        const float w = __expf(d - m);       // uniform per wave
        z += w;
        racc[0] += w * x0.x;  racc[1] += w * x0.y;
        racc[2] += w * x0.z;  racc[3] += w * x0.w;
        racc[4] += w * x1.x;  racc[5] += w * x1.y;
        racc[6] += w * x1.z;  racc[7] += w * x1.w;
    }

#pragma unroll
    for (int j = 0; j < 8; ++j) rp[wave * UNITS + base + j] = racc[j];
    if (lane == 0) { m_sh[wave] = m; z_sh[wave] = z; }
    __syncthreads();

    // fixed-order cross-wave merge (every thread redundantly; deterministic)
    float mb = m_sh[0];
#pragma unroll
    for (int w = 0; w < 8; ++w) mb = fmaxf(mb, m_sh[w]);

    float r = 0.0f, zt = 0.0f;
#pragma unroll
    for (int w = 0; w < 8; ++w) {
        const float sc = __expf(m_sh[w] - mb);   // 0 for empty waves
        r  += rp[w * UNITS + tid] * sc;
        zt += z_sh[w] * sc;
    }
    r = (zt > 0.0f) ? (r / zt) : 0.0f;           // also guards all-empty NaN
    q_star[(size_t)g * QSTAR + UNITS + tid] = r;
}

// ------------------------------------------------------------------
extern "C" void kernel_launch(void* const* d_in, const int* in_sizes, int n_in,
                              void* d_out, int out_size, void* d_ws, size_t ws_size,
                              hipStream_t stream) {
    (void)in_sizes; (void)n_in; (void)out_size; (void)ws_size;

    const float* inputs = (const float*)d_in[0];
    const int*   seg    = (const int*)  d_in[1];
    // d_in[2] = num_graphs scalar (compile-time constant here)
    const float* W_i    = (const float*)d_in[3];
    const float* W_h    = (const float*)d_in[4];
    const float* bias   = (const float*)d_in[5];

    float* q_star = (float*)d_out;                       // [G, 2U] = final output

    char* ws = (char*)d_ws;
    float* h     = (float*)ws;  ws += sizeof(float) * NUM_GRAPHS * UNITS;
    float* c     = (float*)ws;  ws += sizeof(float) * NUM_GRAPHS * UNITS;
    float* gates = (float*)ws;  ws += sizeof(float) * NUM_GRAPHS * GATES;
    int*   starts = (int*)ws;

    // h and c are contiguous -> zero both in one shot; q_star zeroed too
    zero_kernel<<<512, 256, 0, stream>>>(h, NUM_GRAPHS * UNITS * 2);
    zero_kernel<<<512, 256, 0, stream>>>(q_star, NUM_GRAPHS * QSTAR);
    seg_starts_kernel<<<(NUM_GRAPHS + 1 + 255) / 256, 256, 0, stream>>>(seg, starts);

    for (int step = 0; step < STEPS; ++step) {
        gates_gemm_kernel<<<dim3(GATES / 16, NUM_GRAPHS / 16), 32, 0, stream>>>(
            q_star, h, W_i, W_h, gates);
        lstm_cell_kernel<<<(NUM_GRAPHS * UNITS) / 256, 256, 0, stream>>>(
            gates, bias, h, c, q_star);
        attention_kernel<<<NUM_GRAPHS, 256, 0, stream>>>(
            inputs, h, starts, q_star);
    }
}